// TtAttention_5669356830742
// MI455X (gfx1250) — compile-verified
//
#include <hip/hip_runtime.h>

typedef float v2f __attribute__((ext_vector_type(2)));
typedef float v8f __attribute__((ext_vector_type(8)));
typedef unsigned int u32x4 __attribute__((ext_vector_type(4)));
typedef int i32x4 __attribute__((ext_vector_type(4)));
typedef int i32x8 __attribute__((ext_vector_type(8)));

#define HEADS    20
#define HEAD_DIM 64
#define DMODEL   1280
#define SEQ      2048
#define BATCH    2

#if defined(__has_builtin)
#if __has_builtin(__builtin_amdgcn_tensor_load_to_lds) && \
    __has_builtin(__builtin_amdgcn_s_wait_tensorcnt)
#define HAVE_TDM 1
#endif
#endif
#ifndef HAVE_TDM
#define HAVE_TDM 0
#endif

__device__ __forceinline__ v8f wmma_f32(v2f a, v2f b, v8f c) {
  // V_WMMA_F32_16X16X4_F32: D = A(16x4) * B(4x16) + C(16x16), all fp32
  return __builtin_amdgcn_wmma_f32_16x16x4_f32(
      false, a, false, b, (short)0, c, false, false);
}

// LDS byte offset of a shared-memory object: generic pointers to LDS carry the
// 32-bit LDS offset in their low word (flat LDS aperture, ISA 10.2).
#define LDS_OFF(p) ((unsigned)(unsigned long long)(p))

#if HAVE_TDM
// Issue one TDM 2-D tile load: tile_d1 rows of tile_d0 fp32 elements,
// row stride stride0 elements in global memory; LDS destination padded by
// (pad_amount_code+1) DWORDs every 2^(pad_interval_code+1) DWORDs.
__device__ __forceinline__ void tdm_load_2d(
    unsigned lds_byte_off, const void* gptr,
    unsigned tile_d0, unsigned tile_d1, unsigned long long stride0,
    unsigned pad_interval_code, unsigned pad_amount_code) {
  const unsigned long long ga = (unsigned long long)gptr;
  u32x4 g0;
  g0.x = 1u;                                           // count=1, user descriptor
  g0.y = lds_byte_off;                                 // lds_addr
  g0.z = (unsigned)ga;                                 // global_addr[31:0]
  g0.w = (unsigned)((ga >> 32) & 0x1FFFFFFull)         // global_addr[56:32]
       | (2u << 30);                                   // type=2 (image)
  const unsigned w0 = (2u << 16)                       // data_size = 4B
                    | (1u << 20)                       // pad_enable
                    | (pad_interval_code << 22)
                    | (pad_amount_code << 25);
  i32x8 g1;
  g1[0] = (int)w0;
  g1[1] = (int)((tile_d0 & 0xFFFFu) << 16);            // tensor_dim0 = tile_d0
  g1[2] = (int)(((tile_d0 >> 16) & 0xFFFFu) | ((tile_d1 & 0xFFFFu) << 16));
  g1[3] = (int)(((tile_d1 >> 16) & 0xFFFFu) | ((tile_d0 & 0xFFFFu) << 16)); // tile_dim0
  g1[4] = (int)(tile_d1 & 0xFFFFu);                    // tile_dim1 (tile_dim2=0)
  g1[5] = (int)(unsigned)(stride0 & 0xFFFFFFFFull);    // tensor_dim0_stride lo
  g1[6] = (int)(unsigned)((stride0 >> 32) & 0xFFFFull);
  g1[7] = 0;
  i32x4 z4 = {0, 0, 0, 0};
#if __clang_major__ >= 23
  i32x8 z8 = {0, 0, 0, 0, 0, 0, 0, 0};
  __builtin_amdgcn_tensor_load_to_lds(g0, g1, z4, z4, z8, 0);
#else
  __builtin_amdgcn_tensor_load_to_lds(g0, g1, z4, z4, 0);
#endif
}
#endif  // HAVE_TDM

// ---------------------------------------------------------------------------
// Kernel 1: QKV projection.  C[4096,3840] = X[4096,1280] @ Wqkv[1280,3840]
// 64x128 block tile, BK=16, double-buffered LDS staged by the TDM.
// ---------------------------------------------------------------------------
__global__ __launch_bounds__(256) void qkv_gemm_kernel(
    const float* __restrict__ X, const float* __restrict__ Wqkv,
    float* __restrict__ Qb, float* __restrict__ Kb, float* __restrict__ Vb) {
  const int N = 3 * DMODEL;
  const int K = DMODEL;
  __shared__ float As[2][64][20];   // 16 DW row + 4 DW pad (TDM pad 3/3)
  __shared__ float Bs[2][16][132];  // 128 DW row + 4 DW pad (TDM pad 6/3)

  const int t = threadIdx.x;
  const int lane = t & 31;
  const int half = (lane >> 4) & 1;
  const int ln = lane & 15;
  const int hoff = half * 2;
  const int wid = t >> 5;
  const int waveM = wid >> 2;
  const int waveN = wid & 3;
  const int mBase = blockIdx.y * 64;
  const int nBase = blockIdx.x * 128;

  v8f acc[2][2];
#pragma unroll
  for (int i = 0; i < 2; ++i)
#pragma unroll
    for (int j = 0; j < 2; ++j) acc[i][j] = {};

#if !HAVE_TDM
  const int aRow = t >> 2, aCol = (t & 3) * 4;
  const int bRow = t >> 5, bCol = (t & 31) * 4;
#endif

  auto stage = [&](int buf, int k0) {
#if HAVE_TDM
    if (wid == 0) {
      tdm_load_2d(LDS_OFF(&As[buf][0][0]), X + (size_t)mBase * K + k0,
                  16, 64, (unsigned long long)K, 3, 3);
      tdm_load_2d(LDS_OFF(&Bs[buf][0][0]), Wqkv + (size_t)k0 * N + nBase,
                  128, 16, (unsigned long long)N, 6, 3);
    }
#else
    {
      const float4 av = *(const float4*)(X + (size_t)(mBase + aRow) * K + k0 + aCol);
      As[buf][aRow][aCol + 0] = av.x; As[buf][aRow][aCol + 1] = av.y;
      As[buf][aRow][aCol + 2] = av.z; As[buf][aRow][aCol + 3] = av.w;
    }
#pragma unroll
    for (int r = 0; r < 2; ++r) {
      const float4 bv = *(const float4*)(Wqkv + (size_t)(k0 + bRow + r * 8) * N + nBase + bCol);
      Bs[buf][bRow + r * 8][bCol + 0] = bv.x; Bs[buf][bRow + r * 8][bCol + 1] = bv.y;
      Bs[buf][bRow + r * 8][bCol + 2] = bv.z; Bs[buf][bRow + r * 8][bCol + 3] = bv.w;
    }
#endif
  };

  stage(0, 0);
  int buf = 0;
  for (int k0 = 0; k0 < K; k0 += 16) {
    const int cur = buf;
    const bool hasNext = (k0 + 16) < K;
    if (hasNext) stage(1 - cur, k0 + 16);
#if HAVE_TDM
    if (wid == 0) {
      if (hasNext) __builtin_amdgcn_s_wait_tensorcnt(2);   // cur's pair landed
      else         __builtin_amdgcn_s_wait_tensorcnt(0);
    }
#endif
    __syncthreads();

#pragma unroll
    for (int ks = 0; ks < 4; ++ks) {
      const int kk = ks * 4 + hoff;
      v2f a0, a1, b0, b1;
      a0.x = As[cur][waveM * 32 + ln][kk];      a0.y = As[cur][waveM * 32 + ln][kk + 1];
      a1.x = As[cur][waveM * 32 + 16 + ln][kk]; a1.y = As[cur][waveM * 32 + 16 + ln][kk + 1];
      b0.x = Bs[cur][kk][waveN * 32 + ln];      b0.y = Bs[cur][kk + 1][waveN * 32 + ln];
      b1.x = Bs[cur][kk][waveN * 32 + 16 + ln]; b1.y = Bs[cur][kk + 1][waveN * 32 + 16 + ln];
      acc[0][0] = wmma_f32(a0, b0, acc[0][0]);
      acc[0][1] = wmma_f32(a0, b1, acc[0][1]);
      acc[1][0] = wmma_f32(a1, b0, acc[1][0]);
      acc[1][1] = wmma_f32(a1, b1, acc[1][1]);
    }
    __syncthreads();   // everyone done with cur before it is re-staged
    buf = 1 - cur;
  }

  const float qscale = 0.125f;  // 1/sqrt(HEAD_DIM)
#pragma unroll
  for (int i = 0; i < 2; ++i) {
#pragma unroll
    for (int jt = 0; jt < 2; ++jt) {
#pragma unroll
      for (int j = 0; j < 8; ++j) {
        const int m = mBase + waveM * 32 + i * 16 + half * 8 + j;
        const int n = nBase + waveN * 32 + jt * 16 + ln;
        const float v = acc[i][jt][j];
        const int b = m >> 11;
        const int s = m & (SEQ - 1);
        const int part = n / DMODEL;
        const int dm = n % DMODEL;
        const int hh = dm >> 6;
        const int hd = dm & 63;
        const size_t idx = ((size_t)(b * HEADS + hh) * SEQ + s) * HEAD_DIM + hd;
        if (part == 0)      Qb[idx] = v * qscale;
        else if (part == 1) Kb[idx] = v;
        else                Vb[idx] = v;
      }
    }
  }
}

// ---------------------------------------------------------------------------
// Kernel 2: flash attention.  4 waves x 16 query rows; K/V tiles double-
// buffered via TDM; S = 16 WMMAs, online softmax, P via LDS, PV = 16 WMMAs.
// ---------------------------------------------------------------------------
__global__ __launch_bounds__(128) void attn_kernel(
    const float* __restrict__ Qb, const float* __restrict__ Kb,
    const float* __restrict__ Vb, const float* __restrict__ mask,
    float* __restrict__ ctx) {
  __shared__ float Ks[2][16][68];   // 64 DW row + 4 DW pad (TDM pad 5/3)
  __shared__ float Vs[2][16][68];
  __shared__ float Ps[4][16][18];

  const int t = threadIdx.x;
  const int lane = t & 31;
  const int wave = t >> 5;
  const int half = (lane >> 4) & 1;
  const int ln = lane & 15;
  const int hoff = half * 2;

  const int h = blockIdx.y;
  const int b = blockIdx.z;
  const int qrow0 = blockIdx.x * 64 + wave * 16;

  const float* Qp = Qb + ((size_t)(b * HEADS + h) * SEQ + qrow0) * HEAD_DIM;
  const float* Kp = Kb + (size_t)(b * HEADS + h) * SEQ * HEAD_DIM;
  const float* Vp = Vb + (size_t)(b * HEADS + h) * SEQ * HEAD_DIM;
  const float* Mp = mask + (size_t)b * SEQ * SEQ;

  v2f aq[16];
#pragma unroll
  for (int ks = 0; ks < 16; ++ks) {
    const int d = ks * 4 + hoff;
    aq[ks].x = Qp[(size_t)ln * HEAD_DIM + d];
    aq[ks].y = Qp[(size_t)ln * HEAD_DIM + d + 1];
  }

  v8f o[4];
#pragma unroll
  for (int nc = 0; nc < 4; ++nc) o[nc] = {};
  float mrow[8], lrow[8];
#pragma unroll
  for (int j = 0; j < 8; ++j) { mrow[j] = -3.0e38f; lrow[j] = 0.0f; }

  auto stage = [&](int buf, int kt) {
#if HAVE_TDM
    if (wave == 0) {
      tdm_load_2d(LDS_OFF(&Ks[buf][0][0]), Kp + (size_t)kt * 16 * HEAD_DIM,
                  64, 16, (unsigned long long)HEAD_DIM, 5, 3);
      tdm_load_2d(LDS_OFF(&Vs[buf][0][0]), Vp + (size_t)kt * 16 * HEAD_DIM,
                  64, 16, (unsigned long long)HEAD_DIM, 5, 3);
    }
#else
#pragma unroll
    for (int v4 = 0; v4 < 2; ++v4) {
      const int q4 = t + v4 * 128;
      const int r = q4 >> 4;
      const int c = (q4 & 15) * 4;
      const float4 kv = *(const float4*)(Kp + (size_t)(kt * 16 + r) * HEAD_DIM + c);
      Ks[buf][r][c + 0] = kv.x; Ks[buf][r][c + 1] = kv.y;
      Ks[buf][r][c + 2] = kv.z; Ks[buf][r][c + 3] = kv.w;
      const float4 vv = *(const float4*)(Vp + (size_t)(kt * 16 + r) * HEAD_DIM + c);
      Vs[buf][r][c + 0] = vv.x; Vs[buf][r][c + 1] = vv.y;
      Vs[buf][r][c + 2] = vv.z; Vs[buf][r][c + 3] = vv.w;
    }
#endif
  };

  stage(0, 0);
  int buf = 0;
  for (int kt = 0; kt < SEQ / 16; ++kt) {
    const int cur = buf;
    const bool hasNext = (kt + 1) < (SEQ / 16);
    if (hasNext) stage(1 - cur, kt + 1);
#if HAVE_TDM
    if (wave == 0) {
      if (hasNext) __builtin_amdgcn_s_wait_tensorcnt(2);
      else         __builtin_amdgcn_s_wait_tensorcnt(0);
    }
#endif
    __syncthreads();

    // S(16x16) = Qtile (pre-scaled) @ Ktile^T
    v8f c8 = {};
#pragma unroll
    for (int ks = 0; ks < 16; ++ks) {
      const int d = ks * 4 + hoff;
      v2f bk;
      bk.x = Ks[cur][ln][d];
      bk.y = Ks[cur][ln][d + 1];
      c8 = wmma_f32(aq[ks], bk, c8);
    }

    // mask add + online softmax (row = half*8+j, col = ln; 16-lane groups)
    float alpha[8];
#pragma unroll
    for (int j = 0; j < 8; ++j) {
      const int qr = qrow0 + half * 8 + j;
      float s = c8[j] + Mp[(size_t)qr * SEQ + kt * 16 + ln];
      float mx = s;
#pragma unroll
      for (int off = 8; off >= 1; off >>= 1)
        mx = fmaxf(mx, __shfl_xor(mx, off, 32));
      const float newm = fmaxf(mrow[j], mx);
      alpha[j] = __expf(mrow[j] - newm);
      const float p = __expf(s - newm);
      float rs = p;
#pragma unroll
      for (int off = 8; off >= 1; off >>= 1)
        rs += __shfl_xor(rs, off, 32);
      lrow[j] = lrow[j] * alpha[j] + rs;
      mrow[j] = newm;
      Ps[wave][half * 8 + j][ln] = p;
    }
#pragma unroll
    for (int nc = 0; nc < 4; ++nc)
#pragma unroll
      for (int j = 0; j < 8; ++j) o[nc][j] *= alpha[j];

    // P is wave-local: DS ops of one wave are in-order, just drain DScnt.
    asm volatile("s_wait_dscnt 0" ::: "memory");

    // O(16x64) += P(16x16) @ V(16x64)
#pragma unroll
    for (int ks = 0; ks < 4; ++ks) {
      const int kk = ks * 4 + hoff;
      v2f ap;
      ap.x = Ps[wave][ln][kk];
      ap.y = Ps[wave][ln][kk + 1];
#pragma unroll
      for (int nc = 0; nc < 4; ++nc) {
        v2f bv;
        bv.x = Vs[cur][kk][nc * 16 + ln];
        bv.y = Vs[cur][kk + 1][nc * 16 + ln];
        o[nc] = wmma_f32(ap, bv, o[nc]);
      }
    }
    __syncthreads();   // all waves done with cur K/V before re-stage
    buf = 1 - cur;
  }

#pragma unroll
  for (int j = 0; j < 8; ++j) {
    const float inv = 1.0f / lrow[j];
    const int qr = qrow0 + half * 8 + j;
#pragma unroll
    for (int nc = 0; nc < 4; ++nc) {
      ctx[(size_t)(b * SEQ + qr) * DMODEL + h * HEAD_DIM + nc * 16 + ln] =
          o[nc][j] * inv;
    }
  }
}

// ---------------------------------------------------------------------------
// Kernel 3: output projection.  out[4096,1280] = ctx @ Wout + b_out
// ---------------------------------------------------------------------------
__global__ __launch_bounds__(256) void out_gemm_kernel(
    const float* __restrict__ A, const float* __restrict__ Wout,
    const float* __restrict__ bias, float* __restrict__ out) {
  const int N = DMODEL;
  const int K = DMODEL;
  __shared__ float As[2][64][20];
  __shared__ float Bs[2][16][132];

  const int t = threadIdx.x;
  const int lane = t & 31;
  const int half = (lane >> 4) & 1;
  const int ln = lane & 15;
  const int hoff = half * 2;
  const int wid = t >> 5;
  const int waveM = wid >> 2;
  const int waveN = wid & 3;
  const int mBase = blockIdx.y * 64;
  const int nBase = blockIdx.x * 128;

  v8f acc[2][2];
#pragma unroll
  for (int i = 0; i < 2; ++i)
#pragma unroll
    for (int j = 0; j < 2; ++j) acc[i][j] = {};

#if !HAVE_TDM
  const int aRow = t >> 2, aCol = (t & 3) * 4;
  const int bRow = t >> 5, bCol = (t & 31) * 4;
#endif

  auto stage = [&](int buf, int k0) {
#if HAVE_TDM
    if (wid == 0) {
      tdm_load_2d(LDS_OFF(&As[buf][0][0]), A + (size_t)mBase * K + k0,
                  16, 64, (unsigned long long)K, 3, 3);
      tdm_load_2d(LDS_OFF(&Bs[buf][0][0]), Wout + (size_t)k0 * N + nBase,
                  128, 16, (unsigned long long)N, 6, 3);
    }
#else
    {
      const float4 av = *(const float4*)(A + (size_t)(mBase + aRow) * K + k0 + aCol);
      As[buf][aRow][aCol + 0] = av.x; As[buf][aRow][aCol + 1] = av.y;
      As[buf][aRow][aCol + 2] = av.z; As[buf][aRow][aCol + 3] = av.w;
    }
#pragma unroll
    for (int r = 0; r < 2; ++r) {
      const float4 bv = *(const float4*)(Wout + (size_t)(k0 + bRow + r * 8) * N + nBase + bCol);
      Bs[buf][bRow + r * 8][bCol + 0] = bv.x; Bs[buf][bRow + r * 8][bCol + 1] = bv.y;
      Bs[buf][bRow + r * 8][bCol + 2] = bv.z; Bs[buf][bRow + r * 8][bCol + 3] = bv.w;
    }
#endif
  };

  stage(0, 0);
  int buf = 0;
  for (int k0 = 0; k0 < K; k0 += 16) {
    const int cur = buf;
    const bool hasNext = (k0 + 16) < K;
    if (hasNext) stage(1 - cur, k0 + 16);
#if HAVE_TDM
    if (wid == 0) {
      if (hasNext) __builtin_amdgcn_s_wait_tensorcnt(2);
      else         __builtin_amdgcn_s_wait_tensorcnt(0);
    }
#endif
    __syncthreads();

#pragma unroll
    for (int ks = 0; ks < 4; ++ks) {
      const int kk = ks * 4 + hoff;
      v2f a0, a1, b0, b1;
      a0.x = As[cur][waveM * 32 + ln][kk];      a0.y = As[cur][waveM * 32 + ln][kk + 1];
      a1.x = As[cur][waveM * 32 + 16 + ln][kk]; a1.y = As[cur][waveM * 32 + 16 + ln][kk + 1];
      b0.x = Bs[cur][kk][waveN * 32 + ln];      b0.y = Bs[cur][kk + 1][waveN * 32 + ln];
      b1.x = Bs[cur][kk][waveN * 32 + 16 + ln]; b1.y = Bs[cur][kk + 1][waveN * 32 + 16 + ln];
      acc[0][0] = wmma_f32(a0, b0, acc[0][0]);
      acc[0][1] = wmma_f32(a0, b1, acc[0][1]);
      acc[1][0] = wmma_f32(a1, b0, acc[1][0]);
      acc[1][1] = wmma_f32(a1, b1, acc[1][1]);
    }
    __syncthreads();
    buf = 1 - cur;
  }

#pragma unroll
  for (int i = 0; i < 2; ++i) {
#pragma unroll
    for (int jt = 0; jt < 2; ++jt) {
#pragma unroll
      for (int j = 0; j < 8; ++j) {
        const int m = mBase + waveM * 32 + i * 16 + half * 8 + j;
        const int n = nBase + waveN * 32 + jt * 16 + ln;
        out[(size_t)m * DMODEL + n] = acc[i][jt][j] + bias[n];
      }
    }
  }
}

// ---------------------------------------------------------------------------
extern "C" void kernel_launch(void* const* d_in, const int* in_sizes, int n_in,
                              void* d_out, int out_size, void* d_ws, size_t ws_size,
                              hipStream_t stream) {
  const float* x     = (const float*)d_in[0];  // (B,1,S,D)
  const float* mask  = (const float*)d_in[1];  // (B,1,S,S)
  const float* w_qkv = (const float*)d_in[2];  // (D,3D)
  const float* w_out = (const float*)d_in[3];  // (D,D)
  const float* b_out = (const float*)d_in[4];  // (D,)
  float* out = (float*)d_out;

  float* ws = (float*)d_ws;
  const size_t per = (size_t)BATCH * HEADS * SEQ * HEAD_DIM;  // 5,242,880 floats
  float* Qb  = ws;
  float* Kb  = ws + per;
  float* Vb  = ws + 2 * per;
  float* ctx = ws + 3 * per;

  qkv_gemm_kernel<<<dim3(30, 64), 256, 0, stream>>>(x, w_qkv, Qb, Kb, Vb);
  attn_kernel<<<dim3(SEQ / 64, HEADS, BATCH), 128, 0, stream>>>(Qb, Kb, Vb, mask, ctx);
  out_gemm_kernel<<<dim3(10, 64), 256, 0, stream>>>(ctx, w_out, b_out, out);
}